// Measurement_3942779978396
// MI455X (gfx1250) — compile-verified
//
#include <hip/hip_runtime.h>
#include <hip/hip_bf16.h>
#include <math.h>

#define NQ 18
#define BATCH 64
#define DSTATE (1 << NQ)        // 262144
#define HALF   (1 << (NQ - 1))  // 131072
#define SQRT2_2 0.70710678118654752440f

// native clang vectors (HIP's float4 is a class and breaks vector builtins)
typedef float f32x4 __attribute__((ext_vector_type(4)));
typedef float f32x2 __attribute__((ext_vector_type(2)));
typedef int   v4i   __attribute__((ext_vector_type(4)));

// ---------- CDNA5 async global->LDS path (guarded, probe-verified round 2) ----
#if defined(__gfx1250__) && __has_builtin(__builtin_amdgcn_global_load_async_to_lds_b128)
#define USE_ASYNC_LDS 1
typedef __attribute__((address_space(1))) v4i* gptr_v4i;
typedef __attribute__((address_space(3))) v4i* lptr_v4i;
__device__ __forceinline__ void async_copy16(const float* g, float* l) {
  __builtin_amdgcn_global_load_async_to_lds_b128((gptr_v4i)(void*)g,
                                                 (lptr_v4i)(void*)l, 0, 0);
}
__device__ __forceinline__ void wait_async0() {
#if __has_builtin(__builtin_amdgcn_s_wait_asynccnt)
  __builtin_amdgcn_s_wait_asynccnt(0);
#else
  asm volatile("s_wait_asynccnt 0x0" ::: "memory");
#endif
  asm volatile("" ::: "memory");
}
#else
#define USE_ASYNC_LDS 0
#endif

__device__ __forceinline__ unsigned insert_bit(unsigned m, int pb, unsigned j) {
  unsigned low = m & ((1u << pb) - 1u);
  return ((m >> pb) << (pb + 1)) | (j << pb) | low;
}

__device__ __forceinline__ float wave_reduce_add(float v) {
  // wave32 butterfly
  for (int off = 16; off > 0; off >>= 1) v += __shfl_xor(v, off, 32);
  return v;
}

// =====================================================================
// Kernel 0: one-thread prep — hoist trig (Payne–Hanek bloat) out of the
// hot kernels; they just load two uniform floats from workspace.
// =====================================================================
__global__ __launch_bounds__(32) void prep_kernel(const float* __restrict__ angp,
                                                  float* __restrict__ trig) {
  if (threadIdx.x == 0) {
    const float a = *angp;
    trig[0] = cosf(a);    // Re e^{-ia}
    trig[1] = -sinf(a);   // Im e^{-ia}
  }
}

// =====================================================================
// Kernel 1: per-block partial probabilities (deterministic, no atomics)
// grid = BATCH*128 blocks x 256 threads; 1024 m-values per block (4/thread)
// =====================================================================
__global__ __launch_bounds__(256) void probs_kernel(
    const float* __restrict__ xr, const float* __restrict__ xi,
    const float* __restrict__ trig, const int* __restrict__ wirep,
    float* __restrict__ partial) {
  const int b    = blockIdx.x >> 7;
  const int tile = blockIdx.x & 127;
  const int t    = threadIdx.x;
  const int wire = *wirep;
  const int pb   = (NQ - 1) - wire;           // bit position of the measured qubit
  const float er = trig[0], ei = trig[1];     // e^{-i a}
  const size_t base = (size_t)b << NQ;

  float p0 = 0.f, p1 = 0.f;
  const unsigned m0 = (unsigned)tile * 1024u + (unsigned)t * 4u;

  if (pb >= 2) {  // 4 consecutive m are contiguous in the flat state
    const unsigned i0 = insert_bit(m0, pb, 0u);
    const unsigned i1 = i0 | (1u << pb);
    f32x4 r0, q0, r1, q1;
#if USE_ASYNC_LDS
    __shared__ float smem[4096];              // 4 regions x 1024 floats (16 KB)
    async_copy16(xr + base + i0, &smem[0 * 1024 + t * 4]);
    async_copy16(xi + base + i0, &smem[1 * 1024 + t * 4]);
    async_copy16(xr + base + i1, &smem[2 * 1024 + t * 4]);
    async_copy16(xi + base + i1, &smem[3 * 1024 + t * 4]);
    wait_async0();                            // per-wave: each lane reads only its own data
    r0 = *reinterpret_cast<const f32x4*>(&smem[0 * 1024 + t * 4]);
    q0 = *reinterpret_cast<const f32x4*>(&smem[1 * 1024 + t * 4]);
    r1 = *reinterpret_cast<const f32x4*>(&smem[2 * 1024 + t * 4]);
    q1 = *reinterpret_cast<const f32x4*>(&smem[3 * 1024 + t * 4]);
#else
    r0 = *reinterpret_cast<const f32x4*>(xr + base + i0);
    q0 = *reinterpret_cast<const f32x4*>(xi + base + i0);
    r1 = *reinterpret_cast<const f32x4*>(xr + base + i1);
    q1 = *reinterpret_cast<const f32x4*>(xi + base + i1);
#endif
#pragma unroll
    for (int k = 0; k < 4; ++k) {
      const float x0r = r0[k], x0i = q0[k], x1r = r1[k], x1i = q1[k];
      const float tr = er * x1r - ei * x1i;
      const float ti = er * x1i + ei * x1r;
      const float n0r = x0r + tr, n0i = x0i + ti;
      const float n1r = x0r - tr, n1i = x0i - ti;
      p0 += n0r * n0r + n0i * n0i;
      p1 += n1r * n1r + n1i * n1i;
    }
  } else {  // rare generic path (tiny pb): scalar gather
#pragma unroll
    for (int k = 0; k < 4; ++k) {
      const unsigned m = m0 + (unsigned)k;
      const unsigned i0 = insert_bit(m, pb, 0u);
      const unsigned i1 = insert_bit(m, pb, 1u);
      const float x0r = xr[base + i0], x0i = xi[base + i0];
      const float x1r = xr[base + i1], x1i = xi[base + i1];
      const float tr = er * x1r - ei * x1i;
      const float ti = er * x1i + ei * x1r;
      const float n0r = x0r + tr, n0i = x0i + ti;
      const float n1r = x0r - tr, n1i = x0i - ti;
      p0 += n0r * n0r + n0i * n0i;
      p1 += n1r * n1r + n1i * n1i;
    }
  }
  p0 *= 0.5f;  // s^2 = (sqrt2/2)^2
  p1 *= 0.5f;

  // deterministic block reduction: wave32 butterfly + LDS across 8 waves
  p0 = wave_reduce_add(p0);
  p1 = wave_reduce_add(p1);
  __shared__ float red[16];
  const int wav = t >> 5, lane = t & 31;
  if (lane == 0) { red[wav * 2] = p0; red[wav * 2 + 1] = p1; }
  __syncthreads();
  if (t == 0) {
    float s0 = 0.f, s1 = 0.f;
#pragma unroll
    for (int w = 0; w < 8; ++w) { s0 += red[w * 2]; s1 += red[w * 2 + 1]; }
    partial[((size_t)b * 128 + tile) * 2 + 0] = s0;
    partial[((size_t)b * 128 + tile) * 2 + 1] = s1;
  }
}

// =====================================================================
// Kernel 2: exact JAX threefry-2x32-20 Gumbel-max sampling, key(42)
// =====================================================================
__device__ __forceinline__ unsigned rotl32(unsigned x, int r) {
  return (x << r) | (x >> (32 - r));
}
__device__ unsigned threefry_bits(unsigned f) {
  // bits for flat index f of random_bits(key=(0,42), shape=(64,2)):
  // counts = iota(128); x0 = counts[:64], x1 = counts[64:]
  unsigned c0, c1; bool hi;
  if (f < 64u) { c0 = f; c1 = f + 64u; hi = false; }
  else         { c0 = f - 64u; c1 = f; hi = true; }
  const unsigned k0 = 0u, k1 = 42u;
  const unsigned ks[3] = {k0, k1, k0 ^ k1 ^ 0x1BD11BDAu};
  unsigned x0 = c0 + ks[0], x1 = c1 + ks[1];
  const int Re[4] = {13, 15, 26, 6};
  const int Ro[4] = {17, 29, 16, 24};
#pragma unroll
  for (int g = 0; g < 5; ++g) {
    const int* R = (g & 1) ? Ro : Re;
#pragma unroll
    for (int i = 0; i < 4; ++i) { x0 += x1; x1 = rotl32(x1, R[i]); x1 ^= x0; }
    x0 += ks[(g + 1) % 3];
    x1 += ks[(g + 2) % 3] + (unsigned)(g + 1);
  }
  return hi ? x1 : x0;
}
__device__ float gumbel_from_bits(unsigned bits) {
  const float tiny = 1.17549435e-38f;  // finfo(f32).tiny, as in jax.random.gumbel
  const unsigned fb = (bits >> 9) | 0x3f800000u;
  float u = (__uint_as_float(fb) - 1.0f) * (1.0f - tiny) + tiny;
  u = fmaxf(u, tiny);
  return -logf(-logf(u));
}

__global__ __launch_bounds__(64) void sample_kernel(
    const float* __restrict__ partial, float* __restrict__ out_sample,
    int* __restrict__ sel, float* __restrict__ inv) {
  const int b = threadIdx.x;
  if (b >= BATCH) return;
  float p0 = 0.f, p1 = 0.f;
  for (int k = 0; k < 128; ++k) {  // fixed order -> deterministic
    p0 += partial[((size_t)b * 128 + k) * 2 + 0];
    p1 += partial[((size_t)b * 128 + k) * 2 + 1];
  }
  const float g0 = gumbel_from_bits(threefry_bits((unsigned)(2 * b)));
  const float g1 = gumbel_from_bits(threefry_bits((unsigned)(2 * b + 1)));
  const float z0 = logf(p0 + 1e-30f) + g0;
  const float z1 = logf(p1 + 1e-30f) + g1;
  const int s = (z1 > z0) ? 1 : 0;  // argmax, first-wins on ties
  const float ps = s ? p1 : p0;
  sel[b] = s;
  inv[b] = 1.0f / fmaxf(sqrtf(ps), 1e-12f);
  out_sample[b] = (float)s;
}

// =====================================================================
// Kernel 3: gather selected branch, normalize, NT-store interleaved complex
// grid = BATCH*256 blocks x 256 threads; 512 m per block (2/thread)
// =====================================================================
__global__ __launch_bounds__(256) void select_kernel(
    const float* __restrict__ xr, const float* __restrict__ xi,
    const float* __restrict__ trig, const int* __restrict__ wirep,
    const int* __restrict__ sel, const float* __restrict__ inv,
    float* __restrict__ out) {
  const int b    = blockIdx.x >> 8;
  const int tile = blockIdx.x & 255;
  const int t    = threadIdx.x;
  const int wire = *wirep;
  const int pb   = (NQ - 1) - wire;
  const float er = trig[0], ei = trig[1];
  const int s    = sel[b];
  const float sg = s ? -1.f : 1.f;
  const float C  = SQRT2_2 * inv[b];
  const size_t base = (size_t)b << NQ;
  const unsigned m0 = (unsigned)tile * 512u + (unsigned)t * 2u;
  const size_t o = ((size_t)b << (NQ - 1)) + m0;

  float vr0, vi0, vr1, vi1;
  if (pb >= 1) {  // 2 consecutive m contiguous: f32x2 loads, f32x4 store
    const unsigned i0 = insert_bit(m0, pb, 0u);
    const unsigned i1 = i0 | (1u << pb);
    const f32x2 x0r = *reinterpret_cast<const f32x2*>(xr + base + i0);
    const f32x2 x0i = *reinterpret_cast<const f32x2*>(xi + base + i0);
    const f32x2 x1r = *reinterpret_cast<const f32x2*>(xr + base + i1);
    const f32x2 x1i = *reinterpret_cast<const f32x2*>(xi + base + i1);
    float tr = er * x1r.x - ei * x1i.x, ti = er * x1i.x + ei * x1r.x;
    vr0 = C * (x0r.x + sg * tr); vi0 = C * (x0i.x + sg * ti);
    tr = er * x1r.y - ei * x1i.y; ti = er * x1i.y + ei * x1r.y;
    vr1 = C * (x0r.y + sg * tr); vi1 = C * (x0i.y + sg * ti);
  } else {
    const unsigned ia0 = insert_bit(m0, pb, 0u),      ib0 = insert_bit(m0, pb, 1u);
    const unsigned ia1 = insert_bit(m0 + 1u, pb, 0u), ib1 = insert_bit(m0 + 1u, pb, 1u);
    float tr = er * xr[base + ib0] - ei * xi[base + ib0];
    float ti = er * xi[base + ib0] + ei * xr[base + ib0];
    vr0 = C * (xr[base + ia0] + sg * tr); vi0 = C * (xi[base + ia0] + sg * ti);
    tr = er * xr[base + ib1] - ei * xi[base + ib1];
    ti = er * xi[base + ib1] + ei * xr[base + ib1];
    vr1 = C * (xr[base + ia1] + sg * tr); vi1 = C * (xi[base + ia1] + sg * ti);
  }
  f32x4 v; v.x = vr0; v.y = vi0; v.z = vr1; v.w = vi1;
  // NT store: keep the 128MB input resident in the 192MB L2 for this pass
  __builtin_nontemporal_store(v, reinterpret_cast<f32x4*>(out + 2 * o));
}

// =====================================================================
extern "C" void kernel_launch(void* const* d_in, const int* in_sizes, int n_in,
                              void* d_out, int out_size, void* d_ws, size_t ws_size,
                              hipStream_t stream) {
  const float* xr   = (const float*)d_in[0];
  const float* xi   = (const float*)d_in[1];
  const float* ang  = (const float*)d_in[2];
  const int*   wire = (const int*)d_in[3];
  float* out = (float*)d_out;

  float* partial = (float*)d_ws;                              // 64*128*2 f32 = 64 KB
  int*   sel     = (int*)((char*)d_ws + (size_t)BATCH * 128 * 2 * sizeof(float));
  float* inv     = (float*)((char*)sel + BATCH * sizeof(int));
  float* trig    = (float*)((char*)inv + BATCH * sizeof(float));

  float* out_sample = out + (size_t)BATCH * HALF * 2;          // tuple tail: samples

  prep_kernel  <<<1, 32, 0, stream>>>(ang, trig);
  probs_kernel <<<BATCH * 128, 256, 0, stream>>>(xr, xi, trig, wire, partial);
  sample_kernel<<<1, 64, 0, stream>>>(partial, out_sample, sel, inv);
  select_kernel<<<BATCH * 256, 256, 0, stream>>>(xr, xi, trig, wire, sel, inv, out);
}